// MambaMixer_12360915878719
// MI455X (gfx1250) — compile-verified
//
#include <hip/hip_runtime.h>
#include <hip/hip_bf16.h>

// ---------------------------------------------------------------------------
// Mamba2 mixer forward, fp32, MI455X (gfx1250, wave32).
// GEMMs run on V_WMMA_F32_16X16X4_F32 (exact fp32 matrix math).
// ---------------------------------------------------------------------------

constexpr int Bb = 2, S = 2048, HID = 4096;
constexpr int H = 128, P = 64, G = 8, N = 128, KW = 4;
constexpr int D_INNER  = H * P;                    // 8192
constexpr int D_STATE  = G * N;                    // 1024
constexpr int CONV_DIM = D_INNER + 2 * D_STATE;    // 10240
constexpr int IN_OUT   = D_INNER + CONV_DIM + H;   // 18560

constexpr long OUT_OFF_OUT = 0;
constexpr long OUT_OFF_CS  = (long)Bb * S * HID;                       // 16777216
constexpr long OUT_OFF_SSM = OUT_OFF_CS + (long)Bb * CONV_DIM * (KW-1);// +61440

typedef float v2f __attribute__((ext_vector_type(2)));
typedef float v8f __attribute__((ext_vector_type(8)));

// ---------------------------------------------------------------------------
// GEMM: C[M,Nn] = A[M,Kk] * W[Nn,Kk]^T   (row-major, all dims % tile == 0)
// 256 threads = 8 wave32 waves. Block tile 128x128, K chunk 16.
// Wave (wm in 0..3, wn in 0..1) owns a 32x64 C tile = 2x4 accumulators.
// ---------------------------------------------------------------------------
constexpr int BM = 128, BN = 128, KC = 16, LDT = KC + 1;

__global__ __launch_bounds__(256)
void gemm_xwT_f32_wmma(const float* __restrict__ A, const float* __restrict__ W,
                       float* __restrict__ C, int M, int Nn, int Kk) {
  __shared__ float As[BM][LDT];
  __shared__ float Ws[BN][LDT];
  const int tid  = threadIdx.x;
  const int lane = tid & 31;
  const int wave = tid >> 5;
  const int wm   = wave >> 1;              // 0..3 (M dir)
  const int wn   = wave & 1;               // 0..1 (N dir)
  const long bm0 = (long)blockIdx.y * BM;
  const long bn0 = (long)blockIdx.x * BN;

  v8f acc[2][4];
  #pragma unroll
  for (int mi = 0; mi < 2; ++mi)
    #pragma unroll
    for (int ni = 0; ni < 4; ++ni) { v8f z = {}; acc[mi][ni] = z; }

  const int lrow  = lane & 15;             // M (for A frag) / N (for B frag)
  const int khalf = (lane >> 4) << 1;      // K offset 0 or 2 (ISA 16x4 f32 layout)

  for (int kk = 0; kk < Kk; kk += KC) {
    // Cooperative global->LDS: 128x16 floats each for A and W (512 float4 each)
    #pragma unroll
    for (int i = tid; i < (BM * KC) / 4; i += 256) {
      const int row = i >> 2;              // 4 float4 per row
      const int col = (i & 3) * 4;
      const float4 va = *(const float4*)(A + (bm0 + row) * (long)Kk + kk + col);
      As[row][col+0] = va.x; As[row][col+1] = va.y;
      As[row][col+2] = va.z; As[row][col+3] = va.w;
      const float4 vw = *(const float4*)(W + (bn0 + row) * (long)Kk + kk + col);
      Ws[row][col+0] = vw.x; Ws[row][col+1] = vw.y;
      Ws[row][col+2] = vw.z; Ws[row][col+3] = vw.w;
      if (kk + KC < Kk) {                   // pull next K-chunk toward L2/WGP$
        __builtin_prefetch(A + (bm0 + row) * (long)Kk + kk + KC + col, 0, 1);
        __builtin_prefetch(W + (bn0 + row) * (long)Kk + kk + KC + col, 0, 1);
      }
    }
    __syncthreads();

    #pragma unroll
    for (int ks = 0; ks < KC; ks += 4) {
      v2f af[2], bf[4];
      #pragma unroll
      for (int mi = 0; mi < 2; ++mi) {
        const int r = wm * 32 + mi * 16 + lrow;
        af[mi][0] = As[r][ks + khalf];
        af[mi][1] = As[r][ks + khalf + 1];
      }
      #pragma unroll
      for (int ni = 0; ni < 4; ++ni) {
        const int r = wn * 64 + ni * 16 + lrow;
        bf[ni][0] = Ws[r][ks + khalf];
        bf[ni][1] = Ws[r][ks + khalf + 1];
      }
      #pragma unroll
      for (int mi = 0; mi < 2; ++mi)
        #pragma unroll
        for (int ni = 0; ni < 4; ++ni)
          acc[mi][ni] = __builtin_amdgcn_wmma_f32_16x16x4_f32(
              false, af[mi], false, bf[ni], (short)0, acc[mi][ni], false, false);
    }
    __syncthreads();
  }

  // C/D layout: VGPR j -> M = j + 8*(lane>=16), N = lane&15
  #pragma unroll
  for (int mi = 0; mi < 2; ++mi)
    #pragma unroll
    for (int ni = 0; ni < 4; ++ni) {
      const long n = bn0 + wn * 64 + ni * 16 + (lane & 15);
      #pragma unroll
      for (int j = 0; j < 8; ++j) {
        const long m = bm0 + wm * 32 + mi * 16 + j + ((lane >> 4) << 3);
        C[m * (long)Nn + n] = acc[mi][ni][j];
      }
    }
}

// ---------------------------------------------------------------------------
// Depthwise causal conv (K=4) over the xconv slice of proj, + bias + SiLU
// ---------------------------------------------------------------------------
__global__ __launch_bounds__(256)
void conv_silu_kernel(const float* __restrict__ proj, const float* __restrict__ conv_state,
                      const float* __restrict__ conv_w, const float* __restrict__ conv_b,
                      float* __restrict__ xc) {
  const long total = (long)Bb * S * CONV_DIM;
  for (long idx = (long)blockIdx.x * 256 + threadIdx.x; idx < total;
       idx += (long)gridDim.x * 256) {
    const int c = (int)(idx % CONV_DIM);
    const int s = (int)((idx / CONV_DIM) % S);
    const int b = (int)(idx / ((long)CONV_DIM * S));
    float acc = conv_b[c];
    #pragma unroll
    for (int k = 0; k < KW; ++k) {
      const int t = s - (KW - 1) + k;
      float xv;
      if (t >= 0) xv = proj[(long)(b * S + t) * IN_OUT + D_INNER + c];
      else        xv = conv_state[((long)b * CONV_DIM + c) * (KW - 1) + (t + KW - 1)];
      acc = fmaf(xv, conv_w[c * KW + k], acc);
    }
    xc[idx] = acc / (1.0f + __expf(-acc));   // SiLU
  }
}

__global__ __launch_bounds__(256)
void conv_state_out_kernel(const float* __restrict__ proj, float* __restrict__ cs_out) {
  const int idx = blockIdx.x * 256 + threadIdx.x;
  if (idx >= Bb * CONV_DIM * (KW - 1)) return;
  const int j = idx % (KW - 1);
  const int c = (idx / (KW - 1)) % CONV_DIM;
  const int b = idx / ((KW - 1) * CONV_DIM);
  const int t = S - (KW - 1) + j;            // S >= 3, always from fresh x
  cs_out[idx] = proj[(long)(b * S + t) * IN_OUT + D_INNER + c];
}

// ---------------------------------------------------------------------------
// Selective-scan: block per (b,h); 64 threads (one per p); state[N] in VGPRs
// ---------------------------------------------------------------------------
__global__ __launch_bounds__(64)
void scan_kernel(const float* __restrict__ xc, const float* __restrict__ proj,
                 const float* __restrict__ ssm_state0,
                 const float* __restrict__ A_log, const float* __restrict__ Dvec,
                 const float* __restrict__ dt_bias,
                 float* __restrict__ yssm, float* __restrict__ ssm_state_out) {
  const int bh = blockIdx.x;
  const int b  = bh / H;
  const int h  = bh % H;
  const int p  = threadIdx.x;               // 0..63
  const int g  = h / (H / G);               // group index (rep = 16)

  float st[N];
  #pragma unroll
  for (int n = 0; n < N; ++n) st[n] = ssm_state0[(long)bh * P * N + (long)p * N + n];

  const float Ah  = -__expf(A_log[h]);
  const float Dh  = Dvec[h];
  const float dtb = dt_bias[h];

  __shared__ float sB[N], sC[N], sS[2];     // sS = {dt, dA}

  for (int s = 0; s < S; ++s) {
    const long base = (long)(b * S + s) * CONV_DIM;
    #pragma unroll
    for (int n = p; n < N; n += P) {
      sB[n] = xc[base + D_INNER + g * N + n];
      sC[n] = xc[base + D_INNER + D_STATE + g * N + n];
    }
    if (p == 0) {
      const float v  = proj[(long)(b * S + s) * IN_OUT + D_INNER + CONV_DIM + h] + dtb;
      const float dt = (v > 20.0f) ? v : log1pf(__expf(v));   // softplus
      sS[0] = dt;
      sS[1] = __expf(dt * Ah);
    }
    __syncthreads();

    const float xp   = xc[base + h * P + p];
    const float coef = sS[0] * xp;
    const float dA   = sS[1];
    float y = Dh * xp;
    #pragma unroll
    for (int n = 0; n < N; ++n) {
      st[n] = fmaf(st[n], dA, coef * sB[n]);
      y     = fmaf(st[n], sC[n], y);
    }
    yssm[(long)(b * S + s) * D_INNER + h * P + p] = y;
    __syncthreads();
  }

  #pragma unroll
  for (int n = 0; n < N; ++n)
    ssm_state_out[(long)bh * P * N + (long)p * N + n] = st[n];
}

// ---------------------------------------------------------------------------
// Gated group-RMSNorm: block per (b,s); 256 threads; 8 groups of 1024
// ---------------------------------------------------------------------------
__global__ __launch_bounds__(256)
void gnorm_kernel(const float* __restrict__ yssm, const float* __restrict__ proj,
                  const float* __restrict__ norm_w, float* __restrict__ normed) {
  const int bs  = blockIdx.x;
  const int tid = threadIdx.x;
  const long ybase = (long)bs * D_INNER;
  const long pbase = (long)bs * IN_OUT;
  __shared__ float red[256];

  float gated[32];
  float psum[G];
  #pragma unroll
  for (int g = 0; g < G; ++g) {
    float s = 0.0f;
    #pragma unroll
    for (int i = 0; i < 4; ++i) {
      const int d  = g * 1024 + tid * 4 + i;
      const float yv = yssm[ybase + d];
      const float gv = proj[pbase + d];                  // gate slice
      const float gt = yv * (gv / (1.0f + __expf(-gv))); // y * silu(gate)
      gated[g * 4 + i] = gt;
      s = fmaf(gt, gt, s);
    }
    psum[g] = s;
  }

  for (int g = 0; g < G; ++g) {
    red[tid] = psum[g];
    __syncthreads();
    for (int off = 128; off > 0; off >>= 1) {
      if (tid < off) red[tid] += red[tid + off];
      __syncthreads();
    }
    const float rs = rsqrtf(red[0] * (1.0f / 1024.0f) + 1e-6f);
    __syncthreads();
    #pragma unroll
    for (int i = 0; i < 4; ++i) {
      const int d = g * 1024 + tid * 4 + i;
      normed[ybase + d] = gated[g * 4 + i] * rs * norm_w[d];
    }
    __syncthreads();
  }
}

// ---------------------------------------------------------------------------
extern "C" void kernel_launch(void* const* d_in, const int* in_sizes, int n_in,
                              void* d_out, int out_size, void* d_ws, size_t ws_size,
                              hipStream_t stream) {
  (void)in_sizes; (void)n_in; (void)out_size; (void)ws_size;

  const float* hidden     = (const float*)d_in[0];
  const float* conv_state = (const float*)d_in[1];
  const float* ssm_state  = (const float*)d_in[2];
  // d_in[3] = context_lengths (all == S, unused)
  const float* in_proj_w  = (const float*)d_in[4];
  const float* out_proj_w = (const float*)d_in[5];
  const float* conv_w     = (const float*)d_in[6];
  const float* conv_b     = (const float*)d_in[7];
  const float* A_log      = (const float*)d_in[8];
  const float* Dvec       = (const float*)d_in[9];
  const float* dt_bias    = (const float*)d_in[10];
  const float* norm_w     = (const float*)d_in[11];

  float* out = (float*)d_out;
  float* ws  = (float*)d_ws;

  // workspace layout (floats): proj | xc | yssm | normed  (~740 MB total)
  float* proj   = ws;
  float* xc     = proj   + (long)Bb * S * IN_OUT;
  float* yssm   = xc     + (long)Bb * S * CONV_DIM;
  float* normed = yssm   + (long)Bb * S * D_INNER;

  const int Mrows = Bb * S;                 // 4096

  // 1) in_proj: proj = hidden @ in_proj_w^T   (4096 x 18560, K=4096)
  gemm_xwT_f32_wmma<<<dim3(IN_OUT / BN, Mrows / BM), 256, 0, stream>>>(
      hidden, in_proj_w, proj, Mrows, IN_OUT, HID);

  // 2) depthwise conv + SiLU ; new conv_state tail
  {
    const long total = (long)Bb * S * CONV_DIM;
    const int  blks  = (int)(total / 256);  // divisible
    conv_silu_kernel<<<blks, 256, 0, stream>>>(proj, conv_state, conv_w, conv_b, xc);
    const int cs_elems = Bb * CONV_DIM * (KW - 1);
    conv_state_out_kernel<<<(cs_elems + 255) / 256, 256, 0, stream>>>(
        proj, out + OUT_OFF_CS);
  }

  // 3) selective scan (sequential over S per (b,h))
  scan_kernel<<<dim3(Bb * H), 64, 0, stream>>>(
      xc, proj, ssm_state, A_log, Dvec, dt_bias, yssm, out + OUT_OFF_SSM);

  // 4) gated group-RMSNorm
  gnorm_kernel<<<dim3(Bb * S), 256, 0, stream>>>(yssm, proj, norm_w, normed);

  // 5) out_proj: out = normed @ out_proj_w^T  (4096 x 4096, K=8192)
  gemm_xwT_f32_wmma<<<dim3(HID / BN, Mrows / BM), 256, 0, stream>>>(
      normed, out_proj_w, out + OUT_OFF_OUT, Mrows, HID, D_INNER);
}